// wide_basic_21131239097111
// MI455X (gfx1250) — compile-verified
//
#include <hip/hip_runtime.h>
#include <cstdint>

#define B_   32
#define C_   64
#define H_   32
#define W_   32
#define O_   64
#define EPSF 1e-5f

#define PITCH  40     // floats per staged row (16B-aligned rows; img col j at index j+4)
#define ROWS    8     // output rows per workgroup
#define XROWS  10     // staged rows incl. halo
#define OTILE  16     // output channels per workgroup
#define NTHR  256

__device__ __forceinline__ void async_ld_b128(uint32_t lds_off, uint64_t gaddr) {
  asm volatile("global_load_async_to_lds_b128 %0, %1, off"
               :: "v"(lds_off), "v"(gaddr) : "memory");
}
__device__ __forceinline__ void wait_async0() {
  asm volatile("s_wait_asynccnt 0" ::: "memory");
}

// K1: t1 = relu(bn1(x)), elementwise over B*C*H*W
__global__ __launch_bounds__(256) void bnrelu_kernel(
    const float* __restrict__ x,
    const float* __restrict__ g, const float* __restrict__ b,
    const float* __restrict__ m, const float* __restrict__ v,
    float* __restrict__ out)
{
  int idx = blockIdx.x * blockDim.x + threadIdx.x;      // 2,097,152 total
  int c = (idx >> 10) & (C_ - 1);                       // H*W = 1024
  float inv  = g[c] * __frsqrt_rn(v[c] + EPSF);
  float bias = b[c] - m[c] * inv;
  float val  = fmaf(x[idx], inv, bias);
  out[idx] = fmaxf(val, 0.0f);
}

// Adder layer. EPILOGUE 0: y = relu(bn2(adder)) ; EPILOGUE 1: y = adder + resid
template <int EPILOGUE>
__global__ __launch_bounds__(256) void adder_kernel(
    const float* __restrict__ xin,                      // (B,C,H,W)
    const float* __restrict__ w,                        // (O,C,3,3)
    const float* __restrict__ g,  const float* __restrict__ bb,
    const float* __restrict__ mm, const float* __restrict__ vv,
    const float* __restrict__ resid,
    float* __restrict__ out)
{
  __shared__ float xs[C_ * XROWS * PITCH];              // 102,400 B
  __shared__ float ws[OTILE * C_ * 9];                  //  36,864 B

  const int tid = threadIdx.x;
  const int wg  = blockIdx.x;                           // 0..511
  const int b   = wg >> 4;
  const int og  = (wg >> 2) & 3;
  const int rg  = wg & 3;
  const int r0  = rg * ROWS;

  // ---- zero xs (each thread a contiguous 100-float chunk -> ds_store_b128) ----
  {
    const int base = tid * (C_ * XROWS * PITCH / NTHR); // 100 floats per thread
    #pragma unroll
    for (int i = 0; i < C_ * XROWS * PITCH / NTHR; ++i) xs[base + i] = 0.0f;
  }
  __syncthreads();

  // ---- async stage x tile: 640 segments of one (c,row) = 32 floats = 8 x b128 lanes ----
  const uint32_t xs_base = (uint32_t)(uintptr_t)(&xs[0]);
  const uint32_t ws_base = (uint32_t)(uintptr_t)(&ws[0]);
  {
    const int lane8 = tid & 7;
    const int seg0  = tid >> 3;                         // 0..31
    for (int it = 0; it < 20; ++it) {
      int seg = seg0 + it * 32;                         // 0..639
      int c   = seg / XROWS;
      int rr  = seg - c * XROWS;
      int row = r0 - 1 + rr;
      if (row >= 0 && row < H_) {
        uint32_t loff = xs_base +
            (uint32_t)(((c * XROWS + rr) * PITCH + 4 + lane8 * 4) * 4);
        uint64_t gaddr = (uint64_t)(uintptr_t)
            (xin + ((((size_t)b * C_ + c) * H_ + row) * W_) + lane8 * 4);
        async_ld_b128(loff, gaddr);
      }
    }
    // ---- async stage weights: 16 contiguous o-channels = 9216 floats = 36,864 B ----
    const float* wbase = w + (size_t)og * OTILE * C_ * 9;
    #pragma unroll
    for (int it = 0; it < 9; ++it) {
      uint32_t loff  = ws_base + (uint32_t)(tid * 16 + it * 4096);
      uint64_t gaddr = (uint64_t)(uintptr_t)(wbase + tid * 4 + it * 1024);
      async_ld_b128(loff, gaddr);
    }
  }
  wait_async0();
  __syncthreads();

  // ---- compute: thread = (colgroup of 8, row, pair of o-channels) ----
  const int cg = tid & 3;
  const int rl = (tid >> 2) & 7;
  const int op = tid >> 5;                              // 0..7 -> o pair

  float acc0[8], acc1[8];
  #pragma unroll
  for (int p = 0; p < 8; ++p) { acc0[p] = 0.0f; acc1[p] = 0.0f; }

  for (int c = 0; c < C_; ++c) {
    const int xb = (c * XROWS + rl) * PITCH + 3 + cg * 8;   // img col cg*8-1
    float xr[3][10];
    #pragma unroll
    for (int kr = 0; kr < 3; ++kr)
      #pragma unroll
      for (int i = 0; i < 10; ++i)
        xr[kr][i] = xs[xb + kr * PITCH + i];

    const int wb = ((op * 2) * C_ + c) * 9;                 // wave-uniform -> LDS broadcast
    #pragma unroll
    for (int kh = 0; kh < 3; ++kh)
      #pragma unroll
      for (int kw = 0; kw < 3; ++kw) {
        float w0 = ws[wb + kh * 3 + kw];
        float w1 = ws[wb + C_ * 9 + kh * 3 + kw];
        #pragma unroll
        for (int p = 0; p < 8; ++p) {
          acc0[p] -= fabsf(xr[kh][p + kw] - w0);
          acc1[p] -= fabsf(xr[kh][p + kw] - w1);
        }
      }
  }

  // ---- epilogue ----
  const int h  = r0 + rl;
  const int o0 = og * OTILE + op * 2;
  #pragma unroll
  for (int oo = 0; oo < 2; ++oo) {
    const int o = o0 + oo;
    const float* acc = oo ? acc1 : acc0;
    size_t base = ((((size_t)b * O_ + o) * H_) + h) * W_ + cg * 8;
    if (EPILOGUE == 0) {
      float inv  = g[o] * __frsqrt_rn(vv[o] + EPSF);
      float bias = bb[o] - mm[o] * inv;
      #pragma unroll
      for (int p = 0; p < 8; ++p)
        out[base + p] = fmaxf(fmaf(acc[p], inv, bias), 0.0f);
    } else {
      #pragma unroll
      for (int p = 0; p < 8; ++p)
        out[base + p] = acc[p] + resid[base + p];
    }
  }
}

extern "C" void kernel_launch(void* const* d_in, const int* in_sizes, int n_in,
                              void* d_out, int out_size, void* d_ws, size_t ws_size,
                              hipStream_t stream) {
  const float* x  = (const float*)d_in[0];
  const float* w1 = (const float*)d_in[1];
  const float* w2 = (const float*)d_in[2];
  const float* g1 = (const float*)d_in[3];
  const float* b1 = (const float*)d_in[4];
  const float* m1 = (const float*)d_in[5];
  const float* v1 = (const float*)d_in[6];
  const float* g2 = (const float*)d_in[7];
  const float* b2 = (const float*)d_in[8];
  const float* m2 = (const float*)d_in[9];
  const float* v2 = (const float*)d_in[10];

  const size_t tensor_bytes = (size_t)B_ * C_ * H_ * W_ * sizeof(float); // 8 MB
  float* t1 = (float*)d_ws;
  float* t2 = (float*)((char*)d_ws + tensor_bytes);
  float* out = (float*)d_out;

  const int n = B_ * C_ * H_ * W_;
  bnrelu_kernel<<<n / 256, 256, 0, stream>>>(x, g1, b1, m1, v1, t1);
  adder_kernel<0><<<512, 256, 0, stream>>>(t1, w1, g2, b2, m2, v2, nullptr, t2);
  adder_kernel<1><<<512, 256, 0, stream>>>(t2, w2, nullptr, nullptr, nullptr, nullptr, x, out);
}